// SoftToHardQuantize_70480413327499
// MI455X (gfx1250) — compile-verified
//
#include <hip/hip_runtime.h>
#include <stdint.h>

// Soft-to-hard quantize, forward pass only:
//   out[i] = C[argmin_l |x[i] - C[l]|]
// (stop_gradient(hard - y) + y == hard in the forward direction; the softmax
//  path only matters for gradients, so we skip ~134M v_exp_f32 ops and run at
//  the HBM roofline: 64 MB / 23.3 TB/s ~= 2.9 us.)

typedef __attribute__((ext_vector_type(4))) float v4f;

template <int LT>
__global__ __launch_bounds__(256) void stq_nearest_kernel(
    const float* __restrict__ x, const float* __restrict__ C,
    float* __restrict__ out, int n, int Lrt) {
  if constexpr (LT > 0) {
    // ---- Stage the codebook into LDS via gfx1250 async global->LDS DMA ----
    __shared__ float lut[LT];
    if ((int)threadIdx.x < LT) {
      uint32_t ldsoff = (uint32_t)(size_t)(&lut[threadIdx.x]);  // LDS byte offset
      const float* gp = C + threadIdx.x;
      asm volatile("global_load_async_to_lds_b32 %0, %1, off"
                   :
                   : "v"(ldsoff), "v"(gp)
                   : "memory");
    }
    // Only the issuing wave has ASYNCcnt != 0; everyone else passes instantly.
    asm volatile("s_wait_asynccnt 0x0" ::: "memory");
    __syncthreads();

    // Codebook into registers (uniform LDS reads -> broadcast, conflict-free).
    float c[LT];
#pragma unroll
    for (int l = 0; l < LT; ++l) c[l] = lut[l];

    int base = (blockIdx.x * blockDim.x + threadIdx.x) * 4;
    if (base >= n) return;

    if (base + 3 < n) {
      v4f xv = __builtin_nontemporal_load(
          reinterpret_cast<const v4f*>(x + base));  // global_load_b128, TH_NT
      v4f o;
#pragma unroll
      for (int e = 0; e < 4; ++e) {
        float xe = xv[e];
        float bestd = __builtin_inff();
        float bestc = c[0];
#pragma unroll
        for (int l = 0; l < LT; ++l) {
          float d = __builtin_fabsf(xe - c[l]);
          // strict '<' keeps the lowest index on exact ties (matches argmin)
          if (d < bestd) {
            bestd = d;
            bestc = c[l];
          }
        }
        o[e] = bestc;
      }
      __builtin_nontemporal_store(o, reinterpret_cast<v4f*>(out + base));
    } else {
      for (int i = base; i < n; ++i) {
        float xe = x[i];
        float bestd = __builtin_inff();
        float bestc = c[0];
#pragma unroll
        for (int l = 0; l < LT; ++l) {
          float d = __builtin_fabsf(xe - c[l]);
          if (d < bestd) {
            bestd = d;
            bestc = c[l];
          }
        }
        out[i] = bestc;
      }
    }
  } else {
    // Generic runtime-L fallback (never hit by this harness; L == 16).
    int base = (blockIdx.x * blockDim.x + threadIdx.x) * 4;
    if (base >= n) return;
    int end = base + 4 < n ? base + 4 : n;
    for (int i = base; i < end; ++i) {
      float xe = x[i];
      float bestd = __builtin_inff();
      float bestc = C[0];
      for (int l = 0; l < Lrt; ++l) {
        float d = __builtin_fabsf(xe - C[l]);
        if (d < bestd) {
          bestd = d;
          bestc = C[l];
        }
      }
      out[i] = bestc;
    }
  }
}

extern "C" void kernel_launch(void* const* d_in, const int* in_sizes, int n_in,
                              void* d_out, int out_size, void* d_ws,
                              size_t ws_size, hipStream_t stream) {
  const float* x = (const float*)d_in[0];   // input, fp32, N = 8*64*128*128
  const float* C = (const float*)d_in[1];   // codebook [L,1], fp32
  // d_in[2] = sigma (int scalar): unused in the forward value (out == hard).
  float* out = (float*)d_out;

  int n = in_sizes[0];
  int L = (n_in > 1) ? in_sizes[1] : 16;

  int n4 = (n + 3) / 4;
  int blocks = (n4 + 255) / 256;
  if (blocks < 1) blocks = 1;

  if (L == 16) {
    stq_nearest_kernel<16><<<blocks, 256, 0, stream>>>(x, C, out, n, L);
  } else {
    stq_nearest_kernel<0><<<blocks, 256, 0, stream>>>(x, C, out, n, L);
  }
}